// CurveCDLoss_65180423684619
// MI455X (gfx1250) — compile-verified
//
#include <hip/hip_runtime.h>

// ---------------------------------------------------------------------------
// CurveCDLoss for MI455X (gfx1250, wave32)
//   Stage 1: per-cloud KNN(k=8) + 3x3 covariance -> 12-D feature + its |.|^2
//            (scalar VALU, full batch staged in LDS: 48 KB of 320 KB/WGP)
//   Stage 2: chamfer min over 12-D features. Cross term a_i . b_j done with
//            V_WMMA_F32_16X16X4_F32, K=12 = 3 chained WMMAs per 16x16 tile.
//            One wave per 16-row tile, row-min carried in the WMMA C layout.
//            Run twice with operands swapped to get both chamfer directions.
// ---------------------------------------------------------------------------

typedef __attribute__((ext_vector_type(2))) float v2f;   // f32 A/B fragment (16x4 / 4x16)
typedef __attribute__((ext_vector_type(8))) float v8f;   // f32 16x16 C/D fragment

#define BATCH 4
#define NPTS  4096
#define DFEAT 12
#define KNN   8
#define WCURV 0.1f

// ---------------------------------------------------------------------------
__global__ void CurveCD_zero_out(float* out) {
    if (blockIdx.x == 0 && threadIdx.x == 0) out[0] = 0.0f;
}

// ---------------------------------------------------------------------------
// One block = 256 points of one batch; whole batch (4096 pts) staged in LDS.
// grid.x = BATCH * 16
__global__ __launch_bounds__(256)
void CurveCD_curve_kernel(const float* __restrict__ xyz,
                          float* __restrict__ s,     // [B,N,12]
                          float* __restrict__ nrm)   // [B,N] squared norms
{
    __shared__ float sx[NPTS], sy[NPTS], sz[NPTS];   // 48 KB

    const int b = blockIdx.x >> 4;
    const int i = ((blockIdx.x & 15) << 8) + threadIdx.x;
    const float* base = xyz + (size_t)b * NPTS * 3;

    for (int t = threadIdx.x; t < NPTS; t += 256) {
        sx[t] = base[t * 3 + 0];
        sy[t] = base[t * 3 + 1];
        sz[t] = base[t * 3 + 2];
    }
    __syncthreads();

    const float px = sx[i], py = sy[i], pz = sz[i];

    float bd[KNN];
    int   bi[KNN];
#pragma unroll
    for (int t = 0; t < KNN; ++t) { bd[t] = 3.0e38f; bi[t] = 0; }

    // brute-force scan; self (d=0) included, matching top_k(-d, k)
    for (int j = 0; j < NPTS; ++j) {
        const float dx = sx[j] - px, dy = sy[j] - py, dz = sz[j] - pz;
        const float d  = fmaf(dx, dx, fmaf(dy, dy, dz * dz));
        if (d < bd[KNN - 1]) {               // rare after warm-up (~8 ln N hits)
            float vd = d; int vi = j;
#pragma unroll
            for (int t = 0; t < KNN; ++t) {
                if (vd < bd[t]) {
                    const float td = bd[t]; const int ti = bi[t];
                    bd[t] = vd; bi[t] = vi; vd = td; vi = ti;
                }
            }
        }
    }

    // symmetric 3x3 covariance of neighbor offsets
    float xx = 0.f, xy = 0.f, xz = 0.f, yy = 0.f, yz = 0.f, zz = 0.f;
#pragma unroll
    for (int t = 0; t < KNN; ++t) {
        const int j = bi[t];
        const float dx = sx[j] - px, dy = sy[j] - py, dz = sz[j] - pz;
        xx = fmaf(dx, dx, xx); xy = fmaf(dx, dy, xy); xz = fmaf(dx, dz, xz);
        yy = fmaf(dy, dy, yy); yz = fmaf(dy, dz, yz); zz = fmaf(dz, dz, zz);
    }
    const float sc = WCURV / (float)KNN;

    const float row[DFEAT] = { px, py, pz,
                               sc * xx, sc * xy, sc * xz,
                               sc * xy, sc * yy, sc * yz,
                               sc * xz, sc * yz, sc * zz };

    float n2 = 0.f;
    float* orow = s + ((size_t)b * NPTS + i) * DFEAT;
#pragma unroll
    for (int t = 0; t < DFEAT; ++t) {
        orow[t] = row[t];
        n2 = fmaf(row[t], row[t], n2);
    }
    nrm[(size_t)b * NPTS + i] = n2;
}

// ---------------------------------------------------------------------------
// One wave per 16-row tile of sA; loop all 256 column tiles of sB.
// d[i,j] = nA[i] + nB[j] - 2 * (sA_i . sB_j), cross term via 3x WMMA f32 K=4.
// grid: BATCH*256 waves = 32768 threads = 128 blocks x 256.
__global__ __launch_bounds__(256)
void CurveCD_chamfer_min(const float* __restrict__ sA, const float* __restrict__ nA,
                         const float* __restrict__ sB, const float* __restrict__ nB,
                         float* __restrict__ out, float scale)
{
    const int wave  = (blockIdx.x * blockDim.x + threadIdx.x) >> 5;
    const int lane  = threadIdx.x & 31;
    const int b     = wave >> 8;            // 256 row tiles per batch
    const int i0    = (wave & 255) << 4;
    const int lrow  = lane & 15;            // row (A) / col (B) within tile
    const int khalf = (lane >> 4) << 1;     // K element pair: 0,1 or 2,3
    const int mhalf = (lane >> 4) << 3;     // C rows: +0 or +8

    const float* Abase = sA + (size_t)b * NPTS * DFEAT;
    const float* Bbase = sB + (size_t)b * NPTS * DFEAT;

    // A fragments: 16x4 f32 tiles for K = 0..3, 4..7, 8..11 (loaded once)
    v2f afrag[3];
#pragma unroll
    for (int k = 0; k < 3; ++k)
        afrag[k] = *(const v2f*)(Abase + (size_t)(i0 + lrow) * DFEAT + 4 * k + khalf);

    // per-C-register row norms: VGPR r holds row M = r + mhalf
    float n1row[8];
#pragma unroll
    for (int r = 0; r < 8; ++r)
        n1row[r] = nA[(size_t)b * NPTS + i0 + r + mhalf];

    float rowmin[8];
#pragma unroll
    for (int r = 0; r < 8; ++r) rowmin[r] = 3.0e38f;

    for (int j0 = 0; j0 < NPTS; j0 += 16) {
        const float* brow = Bbase + (size_t)(j0 + lrow) * DFEAT;
        __builtin_prefetch(brow + 16 * DFEAT, 0, 1);   // next tile -> global_prefetch_b8

        v8f acc = {0.f, 0.f, 0.f, 0.f, 0.f, 0.f, 0.f, 0.f};
#pragma unroll
        for (int k = 0; k < 3; ++k) {
            const v2f bfrag = *(const v2f*)(brow + 4 * k + khalf);
            // 8 args: (neg_a, A, neg_b, B, c_mod, C, reuse_a, reuse_b)
            acc = __builtin_amdgcn_wmma_f32_16x16x4_f32(
                false, afrag[k], false, bfrag, (short)0, acc, false, false);
        }

        const float n2c = nB[(size_t)b * NPTS + j0 + lrow];
#pragma unroll
        for (int r = 0; r < 8; ++r) {
            const float d = fmaf(-2.0f, acc[r], n1row[r] + n2c);
            rowmin[r] = fminf(rowmin[r], d);
        }
    }

    // min across the 16 lanes holding the tile columns (stays within half-wave)
#pragma unroll
    for (int m = 1; m < 16; m <<= 1) {
#pragma unroll
        for (int r = 0; r < 8; ++r)
            rowmin[r] = fminf(rowmin[r], __shfl_xor(rowmin[r], m, 32));
    }

    float sum = 0.f;
#pragma unroll
    for (int r = 0; r < 8; ++r) sum += rowmin[r];

    if (lrow == 0)                           // lanes 0 and 16: rows 0..7 / 8..15
        atomicAdd(out, sum * scale);
}

// ---------------------------------------------------------------------------
extern "C" void kernel_launch(void* const* d_in, const int* in_sizes, int n_in,
                              void* d_out, int out_size, void* d_ws, size_t ws_size,
                              hipStream_t stream)
{
    (void)in_sizes; (void)n_in; (void)out_size; (void)ws_size;

    const float* xyz1 = (const float*)d_in[0];
    const float* xyz2 = (const float*)d_in[1];
    float* out = (float*)d_out;

    float* ws = (float*)d_ws;
    float* s1 = ws;                                   // B*N*12 floats
    float* s2 = s1 + (size_t)BATCH * NPTS * DFEAT;    // B*N*12 floats
    float* n1 = s2 + (size_t)BATCH * NPTS * DFEAT;    // B*N floats
    float* n2 = n1 + (size_t)BATCH * NPTS;            // B*N floats

    CurveCD_zero_out<<<1, 32, 0, stream>>>(out);

    CurveCD_curve_kernel<<<BATCH * 16, 256, 0, stream>>>(xyz1, s1, n1);
    CurveCD_curve_kernel<<<BATCH * 16, 256, 0, stream>>>(xyz2, s2, n2);

    const float scale = 1.0f / (float)(BATCH * NPTS);
    CurveCD_chamfer_min<<<128, 256, 0, stream>>>(s1, n1, s2, n2, out, scale); // mean(d1)
    CurveCD_chamfer_min<<<128, 256, 0, stream>>>(s2, n2, s1, n1, out, scale); // mean(d2)
}